// DGFCore_43198781063541
// MI455X (gfx1250) — compile-verified
//
#include <hip/hip_runtime.h>
#include <math.h>

// ---------------------------------------------------------------------------
// Types for CDNA5 WMMA f32 16x16x4:
//   A 16x4 f32  -> 2 VGPRs -> v2f ; B 4x16 -> v2f ; C/D 16x16 -> v8f
// ---------------------------------------------------------------------------
typedef __attribute__((ext_vector_type(2))) float v2f;
typedef __attribute__((ext_vector_type(8))) float v8f;

#define HDIM 256

// ---------------------------------------------------------------------------
// WMMA f32 GEMM: C[M,N] = scale * (op(A) @ B) + bias[n]
// One wave per 16x(16*NT) output tile; grid = (N/(16*NT), M/16).
// TRANSA=0: A row-major [M,K], lda=K.  TRANSA=1: A is [K,M], lda=M (A^T @ B).
// B row-major [K,N], ldb=N.  Requires M%16==0, N%(16*NT)==0, K%4==0.
//
// Explicitly software-pipelined: fragments for step k+4 are loaded into
// distinct registers BEFORE the 4 WMMAs of step k are issued, so the
// s_wait before each WMMA group covers loads issued a full step earlier.
// NT independent accumulator chains hide WMMA latency within the group.
// ---------------------------------------------------------------------------
template <int TRANSA, int NT>
__global__ void dgf_wmma_gemm(const float* __restrict__ A,
                              const float* __restrict__ B,
                              const float* __restrict__ bias,
                              float* __restrict__ C,
                              int K, int lda, int ldb, int ldc, float scale) {
    const int lane = threadIdx.x & 31;   // wave32
    const int half = lane >> 4;          // 0/1 : K-pair selector
    const int lm   = lane & 15;
    const int m0 = blockIdx.y << 4;
    const int n0 = blockIdx.x * (16 * NT);

    const float* pA = TRANSA
        ? (A + (size_t)(half * 2) * lda + (m0 + lm))    // a[v] = A[k+2h+v][m]
        : (A + (size_t)(m0 + lm) * lda + half * 2);     // a[v] = A[m][k+2h+v]
    const float* pB = B + (size_t)(half * 2) * ldb + (n0 + lm);
    const size_t stepA = TRANSA ? (size_t)4 * lda : (size_t)4;
    const size_t stepB = (size_t)4 * ldb;

    v8f acc[NT];
#pragma unroll
    for (int j = 0; j < NT; ++j) acc[j] = (v8f){};

    // ---- prologue: load fragments for k = 0 --------------------------------
    v2f a0;
    v2f b0[NT];
    if (TRANSA) { a0.x = pA[0]; a0.y = pA[lda]; }
    else        { a0.x = pA[0]; a0.y = pA[1];   }
#pragma unroll
    for (int j = 0; j < NT; ++j) {
        b0[j].x = pB[16 * j];
        b0[j].y = pB[16 * j + ldb];
    }
    pA += stepA;
    pB += stepB;

    // ---- steady state: prefetch k+4, compute k -----------------------------
#pragma unroll 2
    for (int k = 4; k < K; k += 4) {
        v2f a1;
        v2f b1[NT];
        if (TRANSA) { a1.x = pA[0]; a1.y = pA[lda]; }
        else        { a1.x = pA[0]; a1.y = pA[1];   }
#pragma unroll
        for (int j = 0; j < NT; ++j) {
            b1[j].x = pB[16 * j];
            b1[j].y = pB[16 * j + ldb];
        }
        pA += stepA;
        pB += stepB;

#pragma unroll
        for (int j = 0; j < NT; ++j)
            acc[j] = __builtin_amdgcn_wmma_f32_16x16x4_f32(
                false, a0, false, b0[j], (short)0, acc[j], false, false);

        a0 = a1;
#pragma unroll
        for (int j = 0; j < NT; ++j) b0[j] = b1[j];
    }

    // ---- epilogue: last step -----------------------------------------------
#pragma unroll
    for (int j = 0; j < NT; ++j)
        acc[j] = __builtin_amdgcn_wmma_f32_16x16x4_f32(
            false, a0, false, b0[j], (short)0, acc[j], false, false);

    // ---- store -------------------------------------------------------------
#pragma unroll
    for (int j = 0; j < NT; ++j) {
        const int col = n0 + 16 * j + lm;
        const float bb = bias ? bias[col] : 0.0f;
#pragma unroll
        for (int v = 0; v < 8; ++v) {
            const int row = m0 + v + half * 8;     // D: VGPR v, halves -> M
            C[(size_t)row * ldc + col] = acc[j][v] * scale + bb;
        }
    }
}

// ---------------------------------------------------------------------------
// Row-wise L2 normalize (rows of length 256).  One 256-thread block per row.
// Safe for in == out.
// ---------------------------------------------------------------------------
__global__ void dgf_l2norm(const float* __restrict__ in,
                           float* __restrict__ out) {
    __shared__ float red[HDIM];
    const int row = blockIdx.x;
    const int t = threadIdx.x;
    const float v = in[(size_t)row * HDIM + t];
    red[t] = v * v;
    __syncthreads();
    for (int s = HDIM / 2; s > 0; s >>= 1) {
        if (t < s) red[t] += red[t + s];
        __syncthreads();
    }
    float n = sqrtf(red[0]);
    n = fmaxf(n, 1e-12f);
    out[(size_t)row * HDIM + t] = v / n;
}

// ZM = 0.5*(Z1+Z2) written to two destinations (part_alpha and current).
__global__ void dgf_avg2(const float* __restrict__ z1,
                         const float* __restrict__ z2,
                         float* __restrict__ pa,
                         float* __restrict__ cur, int n) {
    for (int i = blockIdx.x * blockDim.x + threadIdx.x; i < n;
         i += gridDim.x * blockDim.x) {
        const float z = 0.5f * (z1[i] + z2[i]);
        pa[i] = z;
        cur[i] = z;
    }
}

__global__ void dgf_zero(float* __restrict__ p, int n) {
    for (int i = blockIdx.x * blockDim.x + threadIdx.x; i < n;
         i += gridDim.x * blockDim.x)
        p[i] = 0.0f;
}

__global__ void dgf_axpy(float* __restrict__ dst,
                         const float* __restrict__ src, int n) {
    for (int i = blockIdx.x * blockDim.x + threadIdx.x; i < n;
         i += gridDim.x * blockDim.x)
        dst[i] += src[i];
}

// ---------------------------------------------------------------------------
// SpMM scatter: out[src[e]][:] += alpha * val[e] * x[dst[e]][:]
// One 256-thread block per edge; fp32 global atomics.  50000x256 fp32
// accumulator (51 MB) is resident in the 192 MB L2.
// ---------------------------------------------------------------------------
__global__ void dgf_spmm(const int* __restrict__ esrc,
                         const int* __restrict__ edst,
                         const float* __restrict__ eval,
                         const float* __restrict__ x,
                         float* __restrict__ out, float alpha) {
    const int e = blockIdx.x;            // uniform -> scalar loads below
    const int t = threadIdx.x;
    const int s = esrc[e];
    const int d = edst[e];
    const float w = alpha * eval[e];
    atomicAdd(&out[(size_t)s * HDIM + t], w * x[(size_t)d * HDIM + t]);
}

// ---------------------------------------------------------------------------
// Symmetric softmax on a 256x256 matrix, in place.  Single 256-thread block.
//   e = exp(S - max(S));  S = e / (sqrt(rowsum_i + eps) * sqrt(colsum_j + eps))
// ---------------------------------------------------------------------------
__global__ void dgf_symsoftmax(float* __restrict__ S) {
    __shared__ float red[HDIM];
    __shared__ float rows[HDIM];
    __shared__ float cols[HDIM];
    const int t = threadIdx.x;

    float mx = -3.402823466e+38f;
    for (int j = 0; j < HDIM; ++j)
        mx = fmaxf(mx, S[(size_t)t * HDIM + j]);
    red[t] = mx;
    __syncthreads();
    for (int s = HDIM / 2; s > 0; s >>= 1) {
        if (t < s) red[t] = fmaxf(red[t], red[t + s]);
        __syncthreads();
    }
    const float gmax = red[0];
    __syncthreads();

    float rs = 0.0f;
    for (int j = 0; j < HDIM; ++j) {
        const float e = expf(S[(size_t)t * HDIM + j] - gmax);
        S[(size_t)t * HDIM + j] = e;
        rs += e;
    }
    rows[t] = sqrtf(rs + 1e-10f);
    __syncthreads();

    float cs = 0.0f;
    for (int i = 0; i < HDIM; ++i)
        cs += S[(size_t)i * HDIM + t];
    cols[t] = sqrtf(cs + 1e-10f);
    __syncthreads();

    const float rinv = 1.0f / rows[t];
    for (int j = 0; j < HDIM; ++j)
        S[(size_t)t * HDIM + j] = S[(size_t)t * HDIM + j] * rinv / cols[j];
}

// Bm = beta_term * 0.5 * (S1 + S2) = 0.25*(S1+S2)
__global__ void dgf_combine(const float* __restrict__ s1,
                            const float* __restrict__ s2,
                            float* __restrict__ bm, int n) {
    for (int i = blockIdx.x * blockDim.x + threadIdx.x; i < n;
         i += gridDim.x * blockDim.x)
        bm[i] = 0.25f * (s1[i] + s2[i]);
}

// SB = I + Bm + T2 + T3
__global__ void dgf_sumbeta(const float* __restrict__ bm,
                            const float* __restrict__ t2,
                            const float* __restrict__ t3,
                            float* __restrict__ sb, int n) {
    for (int i = blockIdx.x * blockDim.x + threadIdx.x; i < n;
         i += gridDim.x * blockDim.x) {
        const float id = ((i / HDIM) == (i % HDIM)) ? 1.0f : 0.0f;
        sb[i] = id + bm[i] + t2[i] + t3[i];
    }
}

// ---------------------------------------------------------------------------
extern "C" void kernel_launch(void* const* d_in, const int* in_sizes, int n_in,
                              void* d_out, int out_size, void* d_ws, size_t ws_size,
                              hipStream_t stream) {
    const float* X1 = (const float*)d_in[0];
    const float* X2 = (const float*)d_in[1];
    const int* esrc = (const int*)d_in[2];
    const int* edst = (const int*)d_in[3];
    const float* ev = (const float*)d_in[4];
    const float* W1 = (const float*)d_in[5];
    const float* b1 = (const float*)d_in[6];
    const float* W2 = (const float*)d_in[7];
    const float* b2 = (const float*)d_in[8];

    const int H = HDIM;
    const int D1 = in_sizes[5] / H;          // 768
    const int D2 = in_sizes[7] / H;          // 512
    const int Nn = in_sizes[0] / D1;         // 50000
    const int E = in_sizes[2];               // 800000
    const int total = Nn * H;
    const int small = H * H;                 // 65536

    float* out = (float*)d_out;
    float* ZM1 = out;
    float* ZM2 = out + (size_t)total;
    float* HMo = out + 2 * (size_t)total;

    float* ws = (float*)d_ws;
    float* part = ws;                        // [N,H]
    float* cur = ws + (size_t)total;         // [N,H]
    float* nxt = ws + 2 * (size_t)total;     // [N,H]
    float* sm1 = ws + 3 * (size_t)total;     // [H,H]
    float* sm2 = sm1 + small;
    float* Bm = sm2 + small;
    float* T2 = Bm + small;
    float* T3 = T2 + small;
    float* SB = T3 + small;

    const dim3 wave(32);
    const dim3 ew(256);
    const dim3 gBig(H / 64, Nn / 16);        // big GEMMs: 16x64 tiles
    const dim3 gSmall(H / 64, H / 16);       // 256x256 GEMMs
    const float inv_sqrt_n = 1.0f / sqrtf((float)Nn);

    // 1) ZM1 = l2norm(X1@W1 + b1), ZM2 = l2norm(X2@W2 + b2)
    dgf_wmma_gemm<0, 4><<<gBig, wave, 0, stream>>>(X1, W1, b1, ZM1, D1, D1, H, H, 1.0f);
    dgf_wmma_gemm<0, 4><<<gBig, wave, 0, stream>>>(X2, W2, b2, ZM2, D2, D2, H, H, 1.0f);
    dgf_l2norm<<<Nn, ew, 0, stream>>>(ZM1, ZM1);
    dgf_l2norm<<<Nn, ew, 0, stream>>>(ZM2, ZM2);

    // 2) ZM = 0.5*(ZM1+ZM2) -> part_alpha, current
    dgf_avg2<<<2048, ew, 0, stream>>>(ZM1, ZM2, part, cur, total);

    // 3) sim_i = (Z_i^T Z_i)/sqrt(N) ; symmetric softmax ; Bm = 0.25*(S1+S2)
    dgf_wmma_gemm<1, 4><<<gSmall, wave, 0, stream>>>(ZM1, ZM1, nullptr, sm1, Nn, H, H, H, inv_sqrt_n);
    dgf_wmma_gemm<1, 4><<<gSmall, wave, 0, stream>>>(ZM2, ZM2, nullptr, sm2, Nn, H, H, H, inv_sqrt_n);
    dgf_symsoftmax<<<1, ew, 0, stream>>>(sm1);
    dgf_symsoftmax<<<1, ew, 0, stream>>>(sm2);
    dgf_combine<<<256, ew, 0, stream>>>(sm1, sm2, Bm, small);

    // 4) sum_beta = I + Bm + Bm^2 + Bm^3
    dgf_wmma_gemm<0, 4><<<gSmall, wave, 0, stream>>>(Bm, Bm, nullptr, T2, H, H, H, H, 1.0f);
    dgf_wmma_gemm<0, 4><<<gSmall, wave, 0, stream>>>(T2, Bm, nullptr, T3, H, H, H, H, 1.0f);
    dgf_sumbeta<<<256, ew, 0, stream>>>(Bm, T2, T3, SB, small);

    // 5) 4 propagation layers: cur = 0.5*spmm(cur); part += cur
    for (int l = 0; l < 4; ++l) {
        dgf_zero<<<2048, ew, 0, stream>>>(nxt, total);
        dgf_spmm<<<E, ew, 0, stream>>>(esrc, edst, ev, cur, nxt, 0.5f);
        dgf_axpy<<<2048, ew, 0, stream>>>(part, nxt, total);
        float* t = cur; cur = nxt; nxt = t;
    }

    // 6) HM = l2norm((part_alpha @ sum_beta) / 4)
    dgf_wmma_gemm<0, 4><<<gBig, wave, 0, stream>>>(part, SB, nullptr, nxt, H, H, H, H, 0.25f);
    dgf_l2norm<<<Nn, ew, 0, stream>>>(nxt, HMo);
}